// SparseMaxPooling_27762668601797
// MI455X (gfx1250) — compile-verified
//
#include <hip/hip_runtime.h>
#include <cstdint>
#include <math.h>

// Sparse 3x3x3 max pooling for MI455X (gfx1250, wave32).
//
// Roofline: pure gather + masked max-reduce. in_feat = 128MB -> resident in the
// 192MB L2; idx/mask/out (~53MB) are streamed once (NT hints so they don't
// evict in_feat). Ballot-compaction gathers only the ~13.5/27 valid neighbors.
//
// gfx1250-specific paths:
//   - __builtin_amdgcn_ballot_w32      : wave32 mask -> SGPR, scalar loop driver
//   - __builtin_amdgcn_readlane/readfirstlane : scalar gather bases (SADDR loads)
//   - __builtin_nontemporal_load/store : TH=NT cache-policy control

#define K_OFFS 27
#define C_CH   32
#define WAVES_PER_BLOCK 8

__global__ __launch_bounds__(C_CH * WAVES_PER_BLOCK)
void SparseMaxPooling_gfx1250_kernel(const float* __restrict__ in_feat,
                                     const int* __restrict__ nbr_idx,
                                     const unsigned char* __restrict__ nbr_mask,
                                     float* __restrict__ out,
                                     int n_out) {
    const int lane = threadIdx.x & 31;          // channel
    const int wave = threadIdx.x >> 5;          // wave within block
    int row = blockIdx.x * WAVES_PER_BLOCK + wave;
    if (row >= n_out) return;                   // wave-uniform exit

    // row is wave-uniform but the compiler can't prove it; force it into an
    // SGPR so all row-base address math is SALU and loads use SADDR forms.
    row = __builtin_amdgcn_readfirstlane(row);

    const int*           idx_row = nbr_idx  + (size_t)row * K_OFFS;
    const unsigned char* msk_row = nbr_mask + (size_t)row * K_OFFS;

    // Lanes 0..26 fetch this row's (index, mask) pair -- coalesced, streamed
    // once -> non-temporal so they stay out of in_feat's L2 working set.
    int idxl = 0;
    int ml   = 0;
    if (lane < K_OFFS) {
        idxl = __builtin_nontemporal_load(idx_row + lane);
        ml   = (int)__builtin_nontemporal_load(msk_row + lane);
    }

    // Wave-uniform bitmask of valid kernel offsets (lanes >= 27 contribute 0).
    uint32_t m = __builtin_amdgcn_ballot_w32(ml != 0);

    float acc = -INFINITY;

    // Scalar (SGPR) loop over only the valid neighbors (~13.5 of 27 avg).
    // 8-wide unroll: up to 8 independent coalesced 128B gathers in flight
    // before the dependent v_max tree; all guards are uniform s_cbranch.
#define GATHER_ONE(vn)                                                         \
    {                                                                          \
        int k = (int)__builtin_ctz(m); m &= m - 1u;                            \
        uint32_t g = (uint32_t)__builtin_amdgcn_readlane(idxl, k);             \
        vn = (in_feat + (size_t)g * C_CH)[lane];                               \
    }

    while (m) {
        float v0 = acc, v1 = acc, v2 = acc, v3 = acc;
        float v4 = acc, v5 = acc, v6 = acc, v7 = acc;
        GATHER_ONE(v0)
        if (m) GATHER_ONE(v1)
        if (m) GATHER_ONE(v2)
        if (m) GATHER_ONE(v3)
        if (m) GATHER_ONE(v4)
        if (m) GATHER_ONE(v5)
        if (m) GATHER_ONE(v6)
        if (m) GATHER_ONE(v7)
        float a = fmaxf(fmaxf(v0, v1), fmaxf(v2, v3));
        float b = fmaxf(fmaxf(v4, v5), fmaxf(v6, v7));
        acc = fmaxf(acc, fmaxf(a, b));
    }
#undef GATHER_ONE

    // Coalesced 128B store per row; written once -> non-temporal.
    __builtin_nontemporal_store(acc, out + (size_t)row * C_CH + lane);
}

extern "C" void kernel_launch(void* const* d_in, const int* in_sizes, int n_in,
                              void* d_out, int out_size, void* d_ws, size_t ws_size,
                              hipStream_t stream) {
    const float*         in_feat  = (const float*)d_in[0];
    const int*           nbr_idx  = (const int*)d_in[1];
    const unsigned char* nbr_mask = (const unsigned char*)d_in[2];  // JAX bool = 1 byte
    float*               out      = (float*)d_out;

    const int n_out = in_sizes[1] / K_OFFS;   // 200,000

    dim3 block(C_CH * WAVES_PER_BLOCK);       // 256 threads = 8 waves
    dim3 grid((n_out + WAVES_PER_BLOCK - 1) / WAVES_PER_BLOCK);
    SparseMaxPooling_gfx1250_kernel<<<grid, block, 0, stream>>>(
        in_feat, nbr_idx, nbr_mask, out, n_out);
}